// conbination_attention_block_64123861729446
// MI455X (gfx1250) — compile-verified
//
#include <hip/hip_runtime.h>
#include <hip/hip_bf16.h>

// ---------------------------------------------------------------------------
// DANet-style dual attention block, fp32, flash-style PAM.
// B=2, C=32, C8=4, N = 16*16*16 = 4096.
// WMMA: V_WMMA_F32_16X16X4_F32 (K=4 == C/8, exact fit for energy = q@k).
// Softmax row reductions amortized over 8-tile strips; V strips staged
// into LDS cooperatively (4 waves/block share the same batch & strip walk).
// ---------------------------------------------------------------------------

#define BATCH 2
#define CH    32
#define C8    4
#define NTOK  4096
#define NT    (NTOK / 16)   // 256 16-token tiles
#define STRIP 8             // key tiles per softmax-reduction strip
#define SJ    (STRIP * 16)  // 128 key columns per strip

typedef float v2f __attribute__((ext_vector_type(2)));
typedef float v8f __attribute__((ext_vector_type(8)));

__device__ static inline v8f wmma4(v2f a, v2f b, v8f c) {
  // D(16x16,f32) = A(16x4,f32) * B(4x16,f32) + C
  return __builtin_amdgcn_wmma_f32_16x16x4_f32(
      /*neg_a=*/false, a, /*neg_b=*/false, b,
      /*c_mod=*/(short)0, c, /*reuse_a=*/false, /*reuse_b=*/false);
}

// Reductions across the 16 lanes that share a D-layout row (lane-half local:
// xor masks 1,2,4,8 never cross the lane/16 boundary).
__device__ static inline float rowmax16(float v) {
  v = fmaxf(v, __shfl_xor(v, 1, 32));
  v = fmaxf(v, __shfl_xor(v, 2, 32));
  v = fmaxf(v, __shfl_xor(v, 4, 32));
  v = fmaxf(v, __shfl_xor(v, 8, 32));
  return v;
}
__device__ static inline float rowsum16(float v) {
  v += __shfl_xor(v, 1, 32);
  v += __shfl_xor(v, 2, 32);
  v += __shfl_xor(v, 4, 32);
  v += __shfl_xor(v, 8, 32);
  return v;
}

// ---------------------------------------------------------------------------
// Kernel 1: q/k/v 1x1x1 projections via WMMA. One wave per (batch, 16-token
// tile). q,k interleaved [B][N][4] (so PAM loads them as one b64); v [B][32][N].
// A matrix for q/k is packed: rows 0-3 = wq, rows 4-7 = wk, rows 8-15 = 0.
// The row select is branch-free (clamped loads + v_cndmask).
// ---------------------------------------------------------------------------
__global__ void __launch_bounds__(128)
qkv_kernel(const float* __restrict__ x,
           const float* __restrict__ wq, const float* __restrict__ bq,
           const float* __restrict__ wk, const float* __restrict__ bk,
           const float* __restrict__ wv, const float* __restrict__ bv,
           float* __restrict__ q, float* __restrict__ k, float* __restrict__ v) {
  const int lane = threadIdx.x & 31;
  const int wib  = threadIdx.x >> 5;
  const int wg   = blockIdx.x * 4 + wib;        // 0..511
  const int b    = wg >> 8;
  const int n0   = (wg & 255) * 16;
  const int m    = lane & 15, half = lane >> 4;
  const int mq   = m & 3;                       // clamped weight row
  const float* xb = x + b * CH * NTOK;

  // B tiles of x for all 8 K-chunks (reused by q/k and both v row blocks)
  v2f bx[8];
#pragma unroll
  for (int kk = 0; kk < 8; ++kk) {
    bx[kk].x = xb[(4 * kk + 2 * half + 0) * NTOK + n0 + m];
    bx[kk].y = xb[(4 * kk + 2 * half + 1) * NTOK + n0 + m];
  }

  v8f aqk = {0.f, 0.f, 0.f, 0.f, 0.f, 0.f, 0.f, 0.f};
  v8f av0 = {0.f, 0.f, 0.f, 0.f, 0.f, 0.f, 0.f, 0.f};
  v8f av1 = {0.f, 0.f, 0.f, 0.f, 0.f, 0.f, 0.f, 0.f};
#pragma unroll
  for (int kk = 0; kk < 8; ++kk) {
    const int ko = 4 * kk + 2 * half;
    // branch-free packed q/k A tile: unconditional clamped loads + selects
    v2f awq = *(const v2f*)&wq[mq * CH + ko];
    v2f awk = *(const v2f*)&wk[mq * CH + ko];
    v2f a;
    a.x = (m < 4) ? awq.x : ((m < 8) ? awk.x : 0.f);
    a.y = (m < 4) ? awq.y : ((m < 8) ? awk.y : 0.f);
    aqk = wmma4(a, bx[kk], aqk);
    v2f a0 = *(const v2f*)&wv[(m +  0) * CH + ko];
    v2f a1 = *(const v2f*)&wv[(m + 16) * CH + ko];
    av0 = wmma4(a0, bx[kk], av0);
    av1 = wmma4(a1, bx[kk], av1);
  }

  // q/k: D rows 0..7 (held by lane-half 0) are the valid outputs
  if (half == 0) {
    const int n = n0 + m;
#pragma unroll
    for (int r = 0; r < 4; ++r)
      q[(b * NTOK + n) * C8 + r] = aqk[r] + bq[r];
#pragma unroll
    for (int r = 4; r < 8; ++r)
      k[(b * NTOK + n) * C8 + (r - 4)] = aqk[r] + bk[r - 4];
  }
  // v: rows = out channel, cols = token
#pragma unroll
  for (int r = 0; r < 8; ++r) {
    const int c0 = half * 8 + r;
    v[(b * CH + c0) * NTOK + n0 + m]      = av0[r] + bv[c0];
    v[(b * CH + c0 + 16) * NTOK + n0 + m] = av1[r] + bv[c0 + 16];
  }
}

// ---------------------------------------------------------------------------
// Kernel 2: flash PAM. One wave per (batch, 16-query-row tile); 4 waves/block
// share one batch and walk strips in lockstep, so the 32x128 V strip is staged
// into LDS once per block per strip. Online softmax per 8-tile strip.
// ---------------------------------------------------------------------------
__global__ void __launch_bounds__(128)
pam_kernel(const float* __restrict__ q, const float* __restrict__ k,
           const float* __restrict__ v, const float* __restrict__ x,
           const float* __restrict__ gamma_p, float* __restrict__ pam) {
  __shared__ float Vs[CH * SJ];                 // 32x128 V strip (16 KB)
  __shared__ float Pl[4 * STRIP * 256];         // per-wave P staging (32 KB)
  const int lane = threadIdx.x & 31;
  const int wib  = threadIdx.x >> 5;
  const int wg   = blockIdx.x * 4 + wib;        // 0..511
  const int b    = wg >> 8;                     // same for all 4 waves
  const int i0   = (wg & 255) * 16;
  const int m    = lane & 15;
  const int half = lane >> 4;
  float* P = &Pl[wib * STRIP * 256];

  const float* kb = k + b * NTOK * C8;
  const float* vb = v + b * CH * NTOK;

  // cooperative V-staging coordinates: thread -> (channel, 32-col segment)
  const int vc  = threadIdx.x >> 2;             // 0..31
  const int vj0 = (threadIdx.x & 3) * 32;       // 0,32,64,96

  // A = q tile, 16x4: lane holds M=m, K = s + 2*half (one b64 load)
  v2f aq = *(const v2f*)&q[(b * NTOK + i0 + m) * C8 + 2 * half];

  float rmax[8], rsum[8];
  v8f accL = {0.f, 0.f, 0.f, 0.f, 0.f, 0.f, 0.f, 0.f};   // out cols c = 0..15
  v8f accH = {0.f, 0.f, 0.f, 0.f, 0.f, 0.f, 0.f, 0.f};   // out cols c = 16..31
#pragma unroll
  for (int r = 0; r < 8; ++r) { rmax[r] = -3.402823e38f; rsum[r] = 0.f; }

  for (int s0 = 0; s0 < NT; s0 += STRIP) {
    // ---- stage V strip to LDS (block-cooperative, 8 b128 per thread) ----
    {
      const float* src = &vb[vc * NTOK + s0 * 16 + vj0];
      float* dst = &Vs[vc * SJ + vj0];
#pragma unroll
      for (int i = 0; i < 32; i += 4)
        *(float4*)&dst[i] = *(const float4*)&src[i];
    }
    // prefetch next strip's keys (global_prefetch_b8)
    if (s0 + STRIP < NT)
      __builtin_prefetch(&kb[((s0 + STRIP) * 16 + m) * C8], 0, 0);
    __syncthreads();

    // ---- energy tiles for the whole strip (held in registers) ----
    v8f S[STRIP];
#pragma unroll
    for (int t = 0; t < STRIP; ++t) {
      const int j0 = (s0 + t) * 16;
      v2f bk_ = *(const v2f*)&kb[(j0 + m) * C8 + 2 * half];
      v8f zero = {0.f, 0.f, 0.f, 0.f, 0.f, 0.f, 0.f, 0.f};
      S[t] = wmma4(aq, bk_, zero);
    }

    // ---- strip row max: lane-local VALU max, then one butterfly ----
    float mt[8];
#pragma unroll
    for (int r = 0; r < 8; ++r) mt[r] = S[0][r];
#pragma unroll
    for (int t = 1; t < STRIP; ++t)
#pragma unroll
      for (int r = 0; r < 8; ++r) mt[r] = fmaxf(mt[r], S[t][r]);
#pragma unroll
    for (int r = 0; r < 8; ++r) {
      const float mn = fmaxf(rmax[r], rowmax16(mt[r]));
      const float sc = __expf(rmax[r] - mn);
      rsum[r] *= sc;
      accL[r] *= sc;
      accH[r] *= sc;
      rmax[r]  = mn;
    }

    // ---- exponentiate, accumulate lane-local sums, stage P to LDS ----
    float psum[8];
#pragma unroll
    for (int r = 0; r < 8; ++r) psum[r] = 0.f;
#pragma unroll
    for (int t = 0; t < STRIP; ++t)
#pragma unroll
      for (int r = 0; r < 8; ++r) {
        const float p = __expf(S[t][r] - rmax[r]);
        psum[r] += p;
        P[t * 256 + (half * 8 + r) * 16 + m] = p;
      }
    asm volatile("s_wait_dscnt 0" ::: "memory");
#pragma unroll
    for (int r = 0; r < 8; ++r) rsum[r] += rowsum16(psum[r]);

    // ---- acc += P(16x128) @ Vt(128x32), K chunked by 4, V from LDS ----
#pragma unroll
    for (int t = 0; t < STRIP; ++t) {
#pragma unroll
      for (int kk = 0; kk < 4; ++kk) {
        const int jj = t * 16 + 4 * kk + 2 * half;
        v2f ap = *(const v2f*)&P[t * 256 + m * 16 + 4 * kk + 2 * half];
        v2f bl = *(const v2f*)&Vs[(m +  0) * SJ + jj];
        v2f bh = *(const v2f*)&Vs[(m + 16) * SJ + jj];
        accL = wmma4(ap, bl, accL);
        accH = wmma4(ap, bh, accH);
      }
    }
    __syncthreads();   // protect Vs before next strip's staging
  }

  const float gp = gamma_p[0];
#pragma unroll
  for (int r = 0; r < 8; ++r) {
    const int i = i0 + half * 8 + r;            // token
    const float inv = 1.0f / rsum[r];
    const int cL = m, cH = m + 16;              // channels
    pam[(b * CH + cL) * NTOK + i] =
        gp * accL[r] * inv + x[(b * CH + cL) * NTOK + i];
    pam[(b * CH + cH) * NTOK + i] =
        gp * accH[r] * inv + x[(b * CH + cH) * NTOK + i];
  }
}

// ---------------------------------------------------------------------------
// Kernel 3: CAM attention matrix. One block per batch; 4 waves compute the
// 2x2 grid of 16x16 eng blocks via WMMA over K=N; then max-subtraction softmax.
// ---------------------------------------------------------------------------
__global__ void __launch_bounds__(128)
cam_attn_kernel(const float* __restrict__ x, float* __restrict__ cattn) {
  __shared__ float eng[CH * CH];
  const int b    = blockIdx.x;
  const int lane = threadIdx.x & 31;
  const int w    = threadIdx.x >> 5;
  const int rb   = (w >> 1) * 16, cb = (w & 1) * 16;
  const int m    = lane & 15, half = lane >> 4;
  const float* xb = x + b * CH * NTOK;

  v8f acc = {0.f, 0.f, 0.f, 0.f, 0.f, 0.f, 0.f, 0.f};
#pragma unroll 8
  for (int k0 = 0; k0 < NTOK; k0 += 4) {
    v2f a  = *(const v2f*)&xb[(rb + m) * NTOK + k0 + 2 * half];
    v2f bm = *(const v2f*)&xb[(cb + m) * NTOK + k0 + 2 * half];
    acc = wmma4(a, bm, acc);
  }
#pragma unroll
  for (int r = 0; r < 8; ++r)
    eng[(rb + half * 8 + r) * CH + cb + m] = acc[r];
  __syncthreads();

  if (threadIdx.x < CH) {
    const int c = threadIdx.x;
    float e[CH];
    float emin = 3.402823e38f;
#pragma unroll
    for (int d = 0; d < CH; ++d) { e[d] = eng[c * CH + d]; emin = fminf(emin, e[d]); }
    // softmax(rowmax - eng): stable form is exp(emin - e[d]) / sum
    float denom = 0.f;
    float num[CH];
#pragma unroll
    for (int d = 0; d < CH; ++d) { num[d] = __expf(emin - e[d]); denom += num[d]; }
    const float inv = 1.0f / denom;
#pragma unroll
    for (int d = 0; d < CH; ++d)
      cattn[(b * CH + c) * CH + d] = num[d] * inv;
  }
}

// ---------------------------------------------------------------------------
// Kernel 4: fused cam_out + residual + concat + out-conv. One wave per
// (batch, 16-token tile). cam tile computed by WMMA (K=32), staged through
// LDS as B-operand for the K=64 output conv WMMAs.
// ---------------------------------------------------------------------------
__global__ void __launch_bounds__(128)
out_kernel(const float* __restrict__ x, const float* __restrict__ cattn,
           const float* __restrict__ gamma_c, const float* __restrict__ pam,
           const float* __restrict__ wo, const float* __restrict__ bo,
           float* __restrict__ out) {
  __shared__ float camL[4 * 512];               // per-wave 32x16 cam tile
  const int lane = threadIdx.x & 31;
  const int wib  = threadIdx.x >> 5;
  const int wg   = blockIdx.x * 4 + wib;        // 0..511
  const int b    = wg >> 8;
  const int n0   = (wg & 255) * 16;
  const int m    = lane & 15, half = lane >> 4;
  float* cl = &camL[wib * 512];
  const float* xb = x + b * CH * NTOK;
  const float* pb = pam + b * CH * NTOK;
  const float gc  = gamma_c[0];

  // cam = gamma_c * (cattn @ xf) + xf, two 16-row channel blocks
#pragma unroll
  for (int ob = 0; ob < 2; ++ob) {
    v8f acc = {0.f, 0.f, 0.f, 0.f, 0.f, 0.f, 0.f, 0.f};
#pragma unroll
    for (int kk = 0; kk < 8; ++kk) {
      const int kb = 4 * kk + 2 * half;
      v2f a = *(const v2f*)&cattn[(b * CH + ob * 16 + m) * CH + kb];
      v2f bb;
      bb.x = xb[(kb + 0) * NTOK + n0 + m];
      bb.y = xb[(kb + 1) * NTOK + n0 + m];
      acc = wmma4(a, bb, acc);
    }
#pragma unroll
    for (int r = 0; r < 8; ++r) {
      const int c = ob * 16 + half * 8 + r;     // channel (D row)
      cl[c * 16 + m] = gc * acc[r] + xb[c * NTOK + n0 + m];
    }
  }
  asm volatile("s_wait_dscnt 0" ::: "memory");

  // out = wo[:, :32] @ pam + wo[:, 32:] @ cam + bo
#pragma unroll
  for (int ob = 0; ob < 2; ++ob) {
    v8f acc = {0.f, 0.f, 0.f, 0.f, 0.f, 0.f, 0.f, 0.f};
#pragma unroll
    for (int kk = 0; kk < 8; ++kk) {            // pam half of K
      const int kb = 4 * kk + 2 * half;
      v2f a = *(const v2f*)&wo[(ob * 16 + m) * (2 * CH) + kb];
      v2f bb;
      bb.x = pb[(kb + 0) * NTOK + n0 + m];
      bb.y = pb[(kb + 1) * NTOK + n0 + m];
      acc = wmma4(a, bb, acc);
    }
#pragma unroll
    for (int kk = 0; kk < 8; ++kk) {            // cam half of K
      const int kb = 4 * kk + 2 * half;
      v2f a = *(const v2f*)&wo[(ob * 16 + m) * (2 * CH) + CH + kb];
      v2f bb;
      bb.x = cl[(kb + 0) * 16 + m];
      bb.y = cl[(kb + 1) * 16 + m];
      acc = wmma4(a, bb, acc);
    }
#pragma unroll
    for (int r = 0; r < 8; ++r) {
      const int o = ob * 16 + half * 8 + r;     // out channel
      out[(b * CH + o) * NTOK + n0 + m] = acc[r] + bo[o];
    }
  }
}

// ---------------------------------------------------------------------------
extern "C" void kernel_launch(void* const* d_in, const int* in_sizes, int n_in,
                              void* d_out, int out_size, void* d_ws, size_t ws_size,
                              hipStream_t stream) {
  (void)in_sizes; (void)n_in; (void)out_size; (void)ws_size;
  const float* x  = (const float*)d_in[0];
  const float* wq = (const float*)d_in[1];
  const float* bq = (const float*)d_in[2];
  const float* wk = (const float*)d_in[3];
  const float* bk = (const float*)d_in[4];
  const float* wv = (const float*)d_in[5];
  const float* bv = (const float*)d_in[6];
  const float* gp = (const float*)d_in[7];
  const float* gc = (const float*)d_in[8];
  const float* wo = (const float*)d_in[9];
  const float* bo = (const float*)d_in[10];
  float* out = (float*)d_out;

  float* ws    = (float*)d_ws;
  float* q     = ws;                              // [B][N][4]  = 32768
  float* k     = q + BATCH * NTOK * C8;           // [B][N][4]  = 32768
  float* v     = k + BATCH * NTOK * C8;           // [B][32][N] = 262144
  float* pam   = v + BATCH * CH * NTOK;           // 262144
  float* cattn = pam + BATCH * CH * NTOK;         // 2048
  // total: 591872 floats = ~2.26 MB of workspace

  qkv_kernel<<<(BATCH * NT) / 4, 128, 0, stream>>>(x, wq, bq, wk, bk, wv, bv, q, k, v);
  pam_kernel<<<(BATCH * NT) / 4, 128, 0, stream>>>(q, k, v, x, gp, pam);
  cam_attn_kernel<<<BATCH, 128, 0, stream>>>(x, cattn);
  out_kernel<<<(BATCH * NT) / 4, 128, 0, stream>>>(x, cattn, gc, pam, wo, bo, out);
}